// percnn_time_stepping_71133248357056
// MI455X (gfx1250) — compile-verified
//
#include <hip/hip_runtime.h>
#include <cstdint>

#define PDT 0.01f
static constexpr int HW  = 512 * 512;      // 262144 pixels per plane
static constexpr int QPB = HW / 4;         // 65536 float4 per plane
static constexpr int BLK = 256;
static constexpr int BLOCKS_PER_B = QPB / BLK;  // 256

// ---------------- Kernel 1: collapse everything into per-batch bivariate cubics ----
// Final coef layout: coef[(b*2+o)*10 + m], m: 1,u,v,u2,uv,v2,u3,u2v,uv2,v3
// out_u = u + P[b][0](u,v) ; out_v = v + P[b][1](u,v)  (DT folded in)
__global__ __launch_bounds__(32) void percnn_prep(
    const float* __restrict__ nu,
    const float* __restrict__ Wc_in, const float* __restrict__ bc_in,
    const float* __restrict__ Wc_out, const float* __restrict__ bc_out,
    const float* __restrict__ Wd_in, const float* __restrict__ bd_in,
    const float* __restrict__ Wd_out, const float* __restrict__ bd_out,
    const float* __restrict__ Ws_in, const float* __restrict__ bs_in,
    const float* __restrict__ Ws_out, const float* __restrict__ bs_out,
    float* __restrict__ coef)
{
    __shared__ float sgA[4], sgB[4], sgC[4];   // grads[i](s) = gA s^2 + gB s + gC
    __shared__ float sD[2][4];                 // diff[i](s)  = D3 s^3 + ... + D0
    __shared__ float sE[2][10];                // raw src bivariate cubic

    const int t = threadIdx.x;
    if (t < 4) {
        const int i = t;
        float A = 0.f, Bq = 0.f, Cq = 0.f;
        for (int c = 0; c < 16; ++c) {
            const float w0 = Wc_in[(i * 2 + 0) * 16 + c];
            const float w1 = Wc_in[(i * 2 + 1) * 16 + c];
            const float b0 = bc_in[(i * 2 + 0) * 16 + c];
            const float b1 = bc_in[(i * 2 + 1) * 16 + c];
            const float wo = Wc_out[i * 16 + c];
            A  += wo * w0 * w1;
            Bq += wo * (w0 * b1 + b0 * w1);
            Cq += wo * b0 * b1;
        }
        sgA[i] = A;
        sgB[i] = Bq;
        sgC[i] = Cq + bc_out[i];
    } else if (t < 6) {
        const int i = t - 4;
        float d0 = 0.f, d1 = 0.f, d2 = 0.f, d3 = 0.f;
        for (int c = 0; c < 16; ++c) {
            const float w0 = Wd_in[(i * 3 + 0) * 16 + c];
            const float w1 = Wd_in[(i * 3 + 1) * 16 + c];
            const float w2 = Wd_in[(i * 3 + 2) * 16 + c];
            const float b0 = bd_in[(i * 3 + 0) * 16 + c];
            const float b1 = bd_in[(i * 3 + 1) * 16 + c];
            const float b2 = bd_in[(i * 3 + 2) * 16 + c];
            const float wo = Wd_out[i * 16 + c];
            d3 += wo * (w0 * w1 * w2);
            d2 += wo * (w0 * w1 * b2 + w0 * b1 * w2 + b0 * w1 * w2);
            d1 += wo * (w0 * b1 * b2 + b0 * w1 * b2 + b0 * b1 * w2);
            d0 += wo * (b0 * b1 * b2);
        }
        sD[i][0] = d0 + bd_out[i];
        sD[i][1] = d1;
        sD[i][2] = d2;
        sD[i][3] = d3;
    } else if (t < 8) {
        const int o = t - 6;
        float e[10];
        #pragma unroll
        for (int m = 0; m < 10; ++m) e[m] = 0.f;
        for (int c = 0; c < 16; ++c) {
            const float a0 = Ws_in[(0 * 16 + c) * 2 + 0];
            const float b0 = Ws_in[(0 * 16 + c) * 2 + 1];
            const float c0 = bs_in[0 * 16 + c];
            const float a1 = Ws_in[(1 * 16 + c) * 2 + 0];
            const float b1 = Ws_in[(1 * 16 + c) * 2 + 1];
            const float c1 = bs_in[1 * 16 + c];
            const float a2 = Ws_in[(2 * 16 + c) * 2 + 0];
            const float b2 = Ws_in[(2 * 16 + c) * 2 + 1];
            const float c2 = bs_in[2 * 16 + c];
            const float wo = Ws_out[o * 16 + c];
            e[0] += wo * (c0 * c1 * c2);
            e[1] += wo * (a0 * c1 * c2 + c0 * a1 * c2 + c0 * c1 * a2);
            e[2] += wo * (b0 * c1 * c2 + c0 * b1 * c2 + c0 * c1 * b2);
            e[3] += wo * (a0 * a1 * c2 + a0 * c1 * a2 + c0 * a1 * a2);
            e[4] += wo * (a0 * b1 * c2 + a0 * c1 * b2 + b0 * a1 * c2 +
                          b0 * c1 * a2 + c0 * a1 * b2 + c0 * b1 * a2);
            e[5] += wo * (b0 * b1 * c2 + b0 * c1 * b2 + c0 * b1 * b2);
            e[6] += wo * (a0 * a1 * a2);
            e[7] += wo * (a0 * a1 * b2 + a0 * b1 * a2 + b0 * a1 * a2);
            e[8] += wo * (a0 * b1 * b2 + b0 * a1 * b2 + b0 * b1 * a2);
            e[9] += wo * (b0 * b1 * b2);
        }
        e[0] += bs_out[o];
        #pragma unroll
        for (int m = 0; m < 10; ++m) sE[o][m] = e[m];
    }
    __syncthreads();

    // Phase 2: per (batch, output) fold conv (negated), nu*diff, and DT into one cubic.
    if (t < 8) {
        const int b = t >> 1, o = t & 1;
        const float nub = nu[b];
        float f[10];
        if (o == 0) {
            // phi_u: nu*diff_u(u) - [u*gq0(u) + v*gq1(u)] + src0(u,v)
            f[0] = nub * sD[0][0]            + sE[0][0];
            f[1] = nub * sD[0][1] - sgC[0]   + sE[0][1];
            f[2] =                - sgC[1]   + sE[0][2];
            f[3] = nub * sD[0][2] - sgB[0]   + sE[0][3];
            f[4] =                - sgB[1]   + sE[0][4];
            f[5] =                             sE[0][5];
            f[6] = nub * sD[0][3] - sgA[0]   + sE[0][6];
            f[7] =                - sgA[1]   + sE[0][7];
            f[8] =                             sE[0][8];
            f[9] =                             sE[0][9];
        } else {
            // phi_v: nu*diff_v(v) - [u*gq2(v) + v*gq3(v)] + src1(u,v)
            f[0] = nub * sD[1][0]            + sE[1][0];
            f[1] =                - sgC[2]   + sE[1][1];
            f[2] = nub * sD[1][1] - sgC[3]   + sE[1][2];
            f[3] =                             sE[1][3];
            f[4] =                - sgB[2]   + sE[1][4];
            f[5] = nub * sD[1][2] - sgB[3]   + sE[1][5];
            f[6] =                             sE[1][6];
            f[7] =                             sE[1][7];
            f[8] =                - sgA[2]   + sE[1][8];
            f[9] = nub * sD[1][3] - sgA[3]   + sE[1][9];
        }
        #pragma unroll
        for (int m = 0; m < 10; ++m) coef[t * 10 + m] = PDT * f[m];
    }
}

// Horner evaluation of bivariate cubic, coeff order: 1,u,v,u2,uv,v2,u3,u2v,uv2,v3
__device__ __forceinline__ float eval_cubic2(const float* __restrict__ F,
                                             float u, float v) {
    const float t2 = fmaf(F[7], v, F[3]);
    const float t1 = fmaf(fmaf(F[8], v, F[4]), v, F[1]);
    const float t0 = fmaf(fmaf(fmaf(F[9], v, F[5]), v, F[2]), v, F[0]);
    return fmaf(fmaf(fmaf(F[6], u, t2), u, t1), u, t0);
}

// ---------------- Kernel 2: streaming update (memory-bound, async global->LDS path)
__global__ __launch_bounds__(BLK) void percnn_main(
    const float* __restrict__ x, const float* __restrict__ coef,
    float* __restrict__ out)
{
    __shared__ float4 ldsU[BLK];
    __shared__ float4 ldsV[BLK];
    const int tid = threadIdx.x;
    const int b   = blockIdx.x >> 8;                    // 256 blocks per batch
    const int q   = ((blockIdx.x & 255) << 8) | tid;    // float4 index in plane

    const size_t planeU = (size_t)((b * 8 + 7) * 2) * (size_t)HW;  // x[b, T-1, 0]
    const float* up = x + planeU + (size_t)q * 4;
    const float* vp = up + HW;

    // CDNA5 async global->LDS copy (ASYNCcnt); each lane fills its own 16B slot.
    const uint32_t ldsUa = (uint32_t)(uintptr_t)&ldsU[tid];
    const uint32_t ldsVa = (uint32_t)(uintptr_t)&ldsV[tid];
    asm volatile("global_load_async_to_lds_b128 %0, %1, off"
                 :: "v"(ldsUa), "v"((uint64_t)(uintptr_t)up) : "memory");
    asm volatile("global_load_async_to_lds_b128 %0, %1, off"
                 :: "v"(ldsVa), "v"((uint64_t)(uintptr_t)vp) : "memory");

    // 20 uniform coefficients for this block's batch (scalar loads, hidden
    // under the in-flight async copy).
    const float* Fb = coef + (size_t)(b * 2) * 10;
    float F0[10], F1[10];
    #pragma unroll
    for (int m = 0; m < 10; ++m) { F0[m] = Fb[m]; F1[m] = Fb[10 + m]; }

    asm volatile("s_wait_asynccnt 0x0" ::: "memory");
    const float4 u4 = ldsU[tid];
    const float4 v4 = ldsV[tid];

    const float us[4] = {u4.x, u4.y, u4.z, u4.w};
    const float vs[4] = {v4.x, v4.y, v4.z, v4.w};
    float ou[4], ov[4];
    #pragma unroll
    for (int j = 0; j < 4; ++j) {
        const float u = us[j], v = vs[j];
        ou[j] = u + eval_cubic2(F0, u, v);
        ov[j] = v + eval_cubic2(F1, u, v);
    }

    float* po = out + (size_t)(b * 2) * (size_t)HW + (size_t)q * 4;
    *(float4*)(po)      = make_float4(ou[0], ou[1], ou[2], ou[3]);
    *(float4*)(po + HW) = make_float4(ov[0], ov[1], ov[2], ov[3]);
}

extern "C" void kernel_launch(void* const* d_in, const int* in_sizes, int n_in,
                              void* d_out, int out_size, void* d_ws, size_t ws_size,
                              hipStream_t stream) {
    const float* x      = (const float*)d_in[0];
    const float* nu     = (const float*)d_in[1];
    const float* Wc_in  = (const float*)d_in[2];
    const float* bc_in  = (const float*)d_in[3];
    const float* Wc_out = (const float*)d_in[4];
    const float* bc_out = (const float*)d_in[5];
    const float* Wd_in  = (const float*)d_in[6];
    const float* bd_in  = (const float*)d_in[7];
    const float* Wd_out = (const float*)d_in[8];
    const float* bd_out = (const float*)d_in[9];
    const float* Ws_in  = (const float*)d_in[10];
    const float* bs_in  = (const float*)d_in[11];
    const float* Ws_out = (const float*)d_in[12];
    const float* bs_out = (const float*)d_in[13];
    float* coef = (float*)d_ws;  // 80 floats (4 batches x 2 outputs x 10)

    percnn_prep<<<1, 32, 0, stream>>>(nu, Wc_in, bc_in, Wc_out, bc_out,
                                      Wd_in, bd_in, Wd_out, bd_out,
                                      Ws_in, bs_in, Ws_out, bs_out, coef);
    percnn_main<<<4 * BLOCKS_PER_B, BLK, 0, stream>>>(x, coef, (float*)d_out);
}